// PathEncoderBatch_29643864277537
// MI455X (gfx1250) — compile-verified
//
#include <hip/hip_runtime.h>

typedef float v2f __attribute__((ext_vector_type(2)));
typedef float v8f __attribute__((ext_vector_type(8)));

// Problem constants (fixed by the reference harness)
#define NGRAPH   16
#define NNODE    256
#define NEDGE    4096
#define FDIM     64
#define NHEAD    8
#define MAXLEN   2
#define PAIRS    (NNODE * NNODE)          // 65536 (x,y) pairs per graph
#define TILE     16                        // pairs per wave (WMMA M)
#define TILES_PER_GRAPH (PAIRS / TILE)     // 4096
#define WAVES_PER_BLOCK 8
#define BLOCKS_MAIN ((NGRAPH - 1) * TILES_PER_GRAPH / WAVES_PER_BLOCK) // 7680

// Padded B staging in LDS: [l][n][d] with n in 0..15 (n>=8 zeroed), stride 68
#define BSTRIDE 68
#define BSIZE   (MAXLEN * 16 * BSTRIDE)    // 2176 floats

__global__ __launch_bounds__(256)
void path_encoder_wmma_kernel(const float* __restrict__ edge_feat,   // (G*E, D)
                              const float* __restrict__ emb_weight,  // (L*H, D)
                              const int*   __restrict__ dist,        // (G, N, N)
                              const int*   __restrict__ path,        // (G, N, N, L)
                              float*       __restrict__ out)         // (G, N, N, H)
{
    __shared__ float s_B[BSIZE];

    // Stage zero-padded B matrix into LDS: s_B[(l*16+n)*68 + d] = emb[l,n,d] (n<8), else 0
    for (int i = threadIdx.x; i < BSIZE; i += blockDim.x) {
        int l = i / (16 * BSTRIDE);
        int rem = i - l * (16 * BSTRIDE);
        int n = rem / BSTRIDE;
        int d = rem - n * BSTRIDE;
        float v = 0.0f;
        if (n < NHEAD && d < FDIM)
            v = emb_weight[(l * NHEAD + n) * FDIM + d];
        s_B[i] = v;
    }
    __syncthreads();

    const int lane = threadIdx.x & 31;
    const int wave = threadIdx.x >> 5;
    const int tile_global = blockIdx.x * WAVES_PER_BLOCK + wave;
    const int og   = tile_global >> 12;               // output graph 0..14
    const int tbase = (tile_global & (TILES_PER_GRAPH - 1)) * TILE;

    const int r    = lane & 15;                        // row in tile (M for lanes 0-15)
    const int half = lane >> 4;                        // K sub-offset selector
    const int pos  = tbase + r;                        // flattened (x,y) for this lane's row

    // path / dist use graph index og+1; edge_feat uses graph index og
    const int* __restrict__ path_g = path + ((size_t)(og + 1) * PAIRS) * MAXLEN;
    const int* __restrict__ dist_g = dist + (size_t)(og + 1) * PAIRS;

    const int p0 = path_g[(size_t)pos * MAXLEN + 0];
    const int p1 = path_g[(size_t)pos * MAXLEN + 1];
    const float m0 = (p0 >= 0) ? 1.0f : 0.0f;          // branchless zero-row handling
    const float m1 = (p1 >= 0) ? 1.0f : 0.0f;
    const int p0c = (p0 < 0) ? 0 : p0;
    const int p1c = (p1 < 0) ? 0 : p1;

    const float* __restrict__ row0 = edge_feat + ((size_t)og * NEDGE + p0c) * FDIM;
    const float* __restrict__ row1 = edge_feat + ((size_t)og * NEDGE + p1c) * FDIM;

    const int koff = half * 2;                         // lane's K sub-offset (0 or 2)
    const float* __restrict__ sB0 = &s_B[(0 * 16 + r) * BSTRIDE];
    const float* __restrict__ sB1 = &s_B[(1 * 16 + r) * BSTRIDE];

    v8f acc = {};

    // l = 0 : global K = 0..63
    #pragma unroll
    for (int s = 0; s < 16; ++s) {
        const int d0 = s * 4 + koff;
        v2f a = *(const v2f*)(row0 + d0);
        a *= m0;
        v2f b = *(const v2f*)(sB0 + d0);
        acc = __builtin_amdgcn_wmma_f32_16x16x4_f32(
            /*neg_a=*/false, a, /*neg_b=*/false, b,
            /*c_mod=*/(short)0, acc, /*reuse_a=*/false, /*reuse_b=*/false);
    }
    // l = 1 : global K = 64..127
    #pragma unroll
    for (int s = 0; s < 16; ++s) {
        const int d0 = s * 4 + koff;
        v2f a = *(const v2f*)(row1 + d0);
        a *= m1;
        v2f b = *(const v2f*)(sB1 + d0);
        acc = __builtin_amdgcn_wmma_f32_16x16x4_f32(
            false, a, false, b, (short)0, acc, false, false);
    }

    // D layout: lane L holds column N=L&15; VGPR v holds row M=(L>>4)*8+v.
    const int n = r;                                   // head index (column)
    if (n < NHEAD) {
        float* __restrict__ out_g = out + (size_t)og * PAIRS * NHEAD;
        #pragma unroll
        for (int v = 0; v < 8; ++v) {
            const int m = half * 8 + v;
            const int posm = tbase + m;
            int dd = dist_g[posm];
            dd = dd < 1 ? 1 : (dd > MAXLEN ? MAXLEN : dd);
            const float scale = (dd == 1) ? 1.0f : 0.5f;
            out_g[(size_t)posm * NHEAD + n] = acc[v] * scale;
        }
    }
}

// Fill last graph's output region with -1000.0f
__global__ __launch_bounds__(256)
void path_encoder_fill_kernel(float4* __restrict__ out_last)
{
    const size_t i = (size_t)blockIdx.x * blockDim.x + threadIdx.x;
    out_last[i] = make_float4(-1000.0f, -1000.0f, -1000.0f, -1000.0f);
}

extern "C" void kernel_launch(void* const* d_in, const int* in_sizes, int n_in,
                              void* d_out, int out_size, void* d_ws, size_t ws_size,
                              hipStream_t stream)
{
    const float* edge_feat  = (const float*)d_in[0];
    const float* emb_weight = (const float*)d_in[1];
    const int*   dist       = (const int*)d_in[2];
    const int*   path       = (const int*)d_in[3];
    float*       out        = (float*)d_out;

    path_encoder_wmma_kernel<<<BLOCKS_MAIN, 256, 0, stream>>>(
        edge_feat, emb_weight, dist, path, out);

    // Last graph: 65536 * 8 floats = 131072 float4
    float4* out_last = (float4*)(out + (size_t)(NGRAPH - 1) * PAIRS * NHEAD);
    path_encoder_fill_kernel<<<(PAIRS * NHEAD / 4) / 256, 256, 0, stream>>>(out_last);
}